// RoIAlignRotated_28063316312292
// MI455X (gfx1250) — compile-verified
//
#include <hip/hip_runtime.h>
#include <math.h>

// RoIAlignRotated for MI455X (gfx1250).
// feat: (B=2, C=256, H=200, W=304) fp32, NCHW
// rois: (R, 6) fp32 rows of [batch_idx, cx, cy, w, h, angle]
// out : (R, C, 7, 7) fp32
//
// Gather-bound kernel. One block per ROI:
//   Phase 1: threads 0..195 compute the 49*4 sample descriptors ONCE
//            (gather offset + 4 bilinear weights with validity and the
//            1/4 sample-average folded in) and stage them in LDS. These
//            are identical for all 256 channels, so this removes a 256x
//            redundant recomputation of the rotation/clamp/floor math.
//   Phase 2: 256 threads sweep j = c*49 + bin. Per sample: ds_load_b32
//            offset + ds_load_b128 weights + two merged global_load_b64
//            corner pairs + 4 FMAs. Adjacent lanes share a channel and
//            walk adjacent bins (cacheline-friendly gathers); the store
//            out[r*12544 + j] is perfectly coalesced and non-temporal
//            (result is never re-read; keep L2 for the 124MB feature map).

namespace {
constexpr int   kC = 256;
constexpr int   kH = 200;
constexpr int   kW = 304;
constexpr int   kOutH = 7;
constexpr int   kOutW = 7;
constexpr int   kBins = kOutH * kOutW;          // 49
constexpr int   kSamples = kBins * 4;           // 196
constexpr int   kPerRoi = kC * kBins;           // 12544
constexpr float kSpatialScale = 0.25f;
}

__global__ __launch_bounds__(256)
void roi_align_rotated_f32(const float* __restrict__ feat,
                           const float* __restrict__ rois,
                           float* __restrict__ out)
{
    __shared__ int    s_off[kSamples];
    __shared__ float4 s_w[kSamples];

    const int r   = blockIdx.x;
    const int tid = threadIdx.x;

    // ---- block-uniform ROI parameters (compiler scalarizes -> s_load) ----
    const float* rp = rois + (size_t)r * 6;
    const int   bidx  = (int)rp[0];
    const float cx    = rp[1] * kSpatialScale - 0.5f;   // ALIGNED offset
    const float cy    = rp[2] * kSpatialScale - 0.5f;
    const float rw    = rp[3] * kSpatialScale;
    const float rh    = rp[4] * kSpatialScale;
    const float theta = rp[5];                          // CLOCKWISE == false

    // ---- Phase 1: one sample descriptor per thread (196 of 256 active) ----
    if (tid < kSamples) {
        const int bin  = tid >> 2;
        const int samp = tid & 3;
        const int ph   = bin / kOutW;
        const int pw   = bin - ph * kOutW;
        const int sy   = samp >> 1;
        const int sx   = samp & 1;

        const float ct = __cosf(theta);
        const float st = __sinf(theta);

        const float bin_h = rh * (1.0f / kOutH);
        const float bin_w = rw * (1.0f / kOutW);

        const float yy = -0.5f * rh + bin_h * ((float)ph + ((float)sy + 0.5f) * 0.5f);
        const float xx = -0.5f * rw + bin_w * ((float)pw + ((float)sx + 0.5f) * 0.5f);

        // rotate into feature-map coordinates
        float y = yy * ct - xx * st + cy;
        float x = yy * st + xx * ct + cx;

        const bool valid = (y > -1.0f) & (y < (float)kH) &
                           (x > -1.0f) & (x < (float)kW);

        y = fminf(fmaxf(y, 0.0f), (float)(kH - 1));
        x = fminf(fmaxf(x, 0.0f), (float)(kW - 1));

        int y0 = (int)floorf(y);
        int x0 = (int)floorf(x);
        y0 = min(y0, kH - 2);
        x0 = min(x0, kW - 2);

        const float ly = y - (float)y0;
        const float lx = x - (float)x0;
        const float hy = 1.0f - ly;
        const float hx = 1.0f - lx;

        // fold validity mask and the 1/4 sample average into the weights
        const float scale = valid ? 0.25f : 0.0f;

        s_off[tid] = y0 * kW + x0;
        s_w[tid]   = make_float4(scale * hy * hx, scale * hy * lx,
                                 scale * ly * hx, scale * ly * lx);
    }
    __syncthreads();

    const float* __restrict__ fbase =
        feat + (size_t)bidx * (size_t)kC * kH * kW;
    const size_t out_base = (size_t)r * kPerRoi;

    // ---- Phase 2: sweep the 12544 (channel, bin) outputs ----
    for (int j = tid; j < kPerRoi; j += blockDim.x) {
        const int c   = j / kBins;
        const int bin = j - c * kBins;

        const float* __restrict__ cbase = fbase + (size_t)c * (kH * kW);

        float acc = 0.0f;
#pragma unroll
        for (int s = 0; s < 4; ++s) {
            const int    off = s_off[bin * 4 + s];
            const float4 w   = s_w[bin * 4 + s];

            const float* __restrict__ p0 = cbase + off;
            const float v00 = p0[0];
            const float v01 = p0[1];          // merges with v00 -> b64
            const float v10 = p0[kW];
            const float v11 = p0[kW + 1];     // merges with v10 -> b64

            acc += w.x * v00 + w.y * v01 + w.z * v10 + w.w * v11;
        }

        // streamed result: non-temporal store (CDNA5 TH=NT), never re-read.
        __builtin_nontemporal_store(acc, out + out_base + j);
    }
}

extern "C" void kernel_launch(void* const* d_in, const int* in_sizes, int n_in,
                              void* d_out, int out_size, void* d_ws, size_t ws_size,
                              hipStream_t stream) {
    (void)n_in; (void)out_size; (void)d_ws; (void)ws_size;

    const float* feat = (const float*)d_in[0];
    const float* rois = (const float*)d_in[1];
    float*       out  = (float*)d_out;

    const int n_rois = in_sizes[1] / 6;   // 1024

    roi_align_rotated_f32<<<dim3(n_rois), dim3(256), 0, stream>>>(feat, rois, out);
}